// FCOSLoss_9088150798832
// MI455X (gfx1250) — compile-verified
//
#include <hip/hip_runtime.h>

// ---------------------------------------------------------------------------
// FCOS loss for MI455X (gfx1250, wave32).
// Memory-bound (≈60 MB dominant traffic -> ~3us at 23.3 TB/s HBM), so the
// design is: one vectorized (b128) coalesced pass per tensor, deterministic
// two-stage reductions. CDNA5-specific paths exercised:
//   - V_WMMA_F32_16X16X4_F32 (A = ones -> column sums collapse 32 lane
//     accumulators at full f32 precision) in every block reduction
//   - GLOBAL_LOAD_ASYNC_TO_LDS_B32 + s_wait_asynccnt for LDS staging of the
//     per-batch boxes/labels in the assignment kernel
//   - global_prefetch_b8 on the streaming scores pass
// ---------------------------------------------------------------------------

#define L_LOCS 5456
#define BATCH  32
#define NCLS   80
#define NBOX   100
#define INF_A  1000000.0f
#define EPSV   1e-9f

#define FOCAL_BLOCKS 1024
#define LOC_BLOCKS   256

typedef float v2f __attribute__((ext_vector_type(2)));
typedef float v8f __attribute__((ext_vector_type(8)));

// Sum of one float per lane across the wave, through the matrix pipe.
// A = 16x4 ones, B(4x16) holds each lane's value exactly once (second B VGPR
// zero). D[i][j] = sum_k B[k][j]; every D row equals the column-sum row, so
// summing D[0] over all 32 lanes double-counts -> halve.
__device__ __forceinline__ float wave_sum_wmma(float x) {
    v2f a; a[0] = 1.0f; a[1] = 1.0f;
    v2f b; b[0] = x;    b[1] = 0.0f;
    v8f c = {};
    v8f d = __builtin_amdgcn_wmma_f32_16x16x4_f32(
        /*neg_a=*/false, a, /*neg_b=*/false, b,
        /*c_mod=*/(short)0, c, /*reuse_a=*/false, /*reuse_b=*/false);
    float s = d[0];
    s += __shfl_xor(s, 1, 32);
    s += __shfl_xor(s, 2, 32);
    s += __shfl_xor(s, 4, 32);
    s += __shfl_xor(s, 8, 32);
    s += __shfl_xor(s, 16, 32);
    return 0.5f * s;
}

// Block-level sum (blockDim.x multiple of 32, <=256). Result valid on tid 0.
__device__ __forceinline__ float block_sum(float v, float* sw) {
    float ws = wave_sum_wmma(v);
    const int lane = threadIdx.x & 31;
    const int wid  = threadIdx.x >> 5;
    if (lane == 0) sw[wid] = ws;
    __syncthreads();
    float r = 0.0f;
    if (threadIdx.x == 0) {
        const int nw = (int)(blockDim.x >> 5);
        for (int i = 0; i < nw; ++i) r += sw[i];
    }
    __syncthreads();   // allow buffer reuse by subsequent calls
    return r;
}

// Stable sigmoid focal-loss term for one logit.
__device__ __forceinline__ float focal_term(float x, bool pos) {
    const float l1p    = log1pf(__expf(-fabsf(x)));
    const float logp   = fminf(x, 0.0f)  - l1p;   // log sigmoid(x)
    const float log1mp = fminf(-x, 0.0f) - l1p;   // log sigmoid(-x)
    const float p      = 1.0f / (1.0f + __expf(-x));
    return pos ? (-0.25f * (1.0f - p) * (1.0f - p) * logp)
               : (-0.75f * p * p * log1mp);
}

// ---------------------------------------------------------------------------
// Kernel A: per-(batch, location) target assignment.
// Boxes + labels staged into LDS via the CDNA5 async-to-LDS path.
// Stable-descending-area argmin == sequential scan with strict '<' on area.
// ---------------------------------------------------------------------------
__global__ void assign_kernel(const float* __restrict__ boxes,
                              const int*   __restrict__ labels,
                              const float* __restrict__ locs,
                              int*   __restrict__ lab_out,
                              float4* __restrict__ bt_out) {
    __shared__ float sb[NBOX * 4];
    __shared__ int   slab[NBOX];
    const int b = blockIdx.y;

    // Async-stage 500 dwords (400 box floats + 100 labels) into LDS.
    // Per-lane: copies a dword from the global address in src to the LDS address in dst; tracked by ASYNCcnt.
    for (int i = threadIdx.x; i < NBOX * 5; i += blockDim.x) {
        const bool isbox = (i < NBOX * 4);
        const unsigned long long src = isbox
            ? (unsigned long long)(uintptr_t)(boxes + (size_t)b * NBOX * 4 + i)
            : (unsigned long long)(uintptr_t)(labels + b * NBOX + (i - NBOX * 4));
        const unsigned dst = isbox
            ? (unsigned)(uintptr_t)&sb[i]
            : (unsigned)(uintptr_t)&slab[i - NBOX * 4];
        asm volatile("global_load_async_to_lds_b32 %0, %1, off"
                     :: "v"(dst), "v"(src) : "memory");
    }
    asm volatile("s_wait_asynccnt 0x0" ::: "memory");
    __syncthreads();

    const int l = blockIdx.x * blockDim.x + threadIdx.x;
    if (l >= L_LOCS) return;

    const float lx = locs[2 * l + 0];
    const float ly = locs[2 * l + 1];

    // FPN level constants: stride*RADIUS and size range, by location index.
    float s, lo, hi;
    if (l < 4096)      { s = 0.0234375f; lo = 0.00f; hi = 0.08f; }
    else if (l < 5120) { s = 0.046875f;  lo = 0.08f; hi = 0.16f; }
    else if (l < 5376) { s = 0.09375f;   lo = 0.16f; hi = 0.32f; }
    else if (l < 5440) { s = 0.1875f;    lo = 0.32f; hi = 0.64f; }
    else               { s = 0.375f;     lo = 0.64f; hi = 1.00f; }

    float best_area = INF_A;
    int   best_lab  = 0;
    float bl = 0.0f, bt = 0.0f, br = 0.0f, bb = 0.0f;

    for (int m = 0; m < NBOX; ++m) {
        const float x0 = sb[4 * m + 0], y0 = sb[4 * m + 1];
        const float x1 = sb[4 * m + 2], y1 = sb[4 * m + 3];
        const float area = (x1 - x0) * (y1 - y0);
        const float dl = lx - x0, dt = ly - y0, dr = x1 - lx, db = y1 - ly;
        const float cx = (x0 + x1) * 0.5f, cy = (y0 + y1) * 0.5f;
        const float xmin = fmaxf(cx - s, x0), ymin = fmaxf(cy - s, y0);
        const float xmax = fminf(cx + s, x1), ymax = fminf(cy + s, y1);
        const float ins = fminf(fminf(lx - xmin, ly - ymin),
                                fminf(xmax - lx, ymax - ly));
        const float mx = fmaxf(fmaxf(dl, dt), fmaxf(dr, db));
        const bool valid = (ins > 0.0f) && (mx >= lo) && (mx <= hi);
        if (valid && (area < best_area)) {
            best_area = area; best_lab = slab[m];
            bl = dl; bt = dt; br = dr; bb = db;
        }
    }

    const int o = b * L_LOCS + l;
    lab_out[o] = (best_area >= INF_A) ? 0 : best_lab;
    bt_out[o] = make_float4(bl, bt, br, bb);
}

// ---------------------------------------------------------------------------
// Kernel B: sigmoid focal loss summed over all (b, l, c). b128 coalesced
// grid-stride pass over the 56 MB score tensor (4 consecutive classes of the
// same location per thread, since 80 % 4 == 0); WMMA block reduction.
// ---------------------------------------------------------------------------
__global__ void focal_kernel(const float4* __restrict__ scores4,
                             const int*    __restrict__ lab,
                             float* __restrict__ partials,
                             int totalq) {
    __shared__ float sw[8];
    const int stride = gridDim.x * blockDim.x;
    float acc = 0.0f;
    for (int q = blockIdx.x * blockDim.x + threadIdx.x; q < totalq; q += stride) {
        if (q + stride < totalq) __builtin_prefetch(&scores4[q + stride], 0, 1);
        const float4 v  = scores4[q];
        const int   row = q / (NCLS / 4);              // (b, l) row
        const int   c1  = (q - row * (NCLS / 4)) * 4 + 1;  // 1-based class of v.x
        const int   t   = lab[row];
        acc += focal_term(v.x, t == c1);
        acc += focal_term(v.y, t == c1 + 1);
        acc += focal_term(v.z, t == c1 + 2);
        acc += focal_term(v.w, t == c1 + 3);
    }
    const float bs = block_sum(acc, sw);
    if (threadIdx.x == 0) partials[blockIdx.x] = bs;
}

// ---------------------------------------------------------------------------
// Kernel C: per-location centerness target, DIoU, centerness BCE.
// Emits 4 partial sums per block: {n_pos, sum(diou*w), sum(w), sum(bce*pos)}.
// ---------------------------------------------------------------------------
__global__ void loc_kernel(const float4* __restrict__ plocs4,
                           const float*  __restrict__ pctr,
                           const int*    __restrict__ lab,
                           const float4* __restrict__ bt4,
                           float* __restrict__ partials,
                           int n) {
    __shared__ float sw[8];
    const int stride = gridDim.x * blockDim.x;
    float a_np = 0.0f, a_dw = 0.0f, a_w = 0.0f, a_bc = 0.0f;

    for (int i = blockIdx.x * blockDim.x + threadIdx.x; i < n; i += stride) {
        const float posf = (lab[i] > 0) ? 1.0f : 0.0f;
        const float4 t4 = bt4[i];
        const float tl = t4.x, tt = t4.y, tr = t4.z, tb = t4.w;

        const float craw = (fminf(tl, tr) / fmaxf(fmaxf(tl, tr), EPSV)) *
                           (fminf(tt, tb) / fmaxf(fmaxf(tt, tb), EPSV));
        const float ctr = (posf > 0.0f) ? sqrtf(fmaxf(craw, EPSV)) : 0.0f;

        const float4 p4 = plocs4[i];
        const float px1 = p4.x, py1 = p4.y, px2 = p4.z, py2 = p4.w;
        const float iw = fmaxf(fminf(px2, tr) - fmaxf(px1, tl), 0.0f);
        const float ih = fmaxf(fminf(py2, tb) - fmaxf(py1, tt), 0.0f);
        const float inter = iw * ih;
        const float uni = (px2 - px1) * (py2 - py1) + (tr - tl) * (tb - tt) - inter;
        const float iou = inter / fmaxf(uni, EPSV);
        const float dx = px1 + px2 - tl - tr;
        const float dy = py1 + py2 - tt - tb;
        const float d2 = (dx * dx + dy * dy) * 0.25f;
        const float ew = fmaxf(px2, tr) - fminf(px1, tl);
        const float eh = fmaxf(py2, tb) - fminf(py1, tt);
        const float diou = 1.0f - iou + d2 / fmaxf(ew * ew + eh * eh, EPSV);

        const float x = pctr[i];
        const float bce = fmaxf(x, 0.0f) - x * ctr + log1pf(__expf(-fabsf(x)));

        a_np += posf;
        a_dw += diou * ctr;   // w = ctr * posf, and ctr is already masked
        a_w  += ctr;
        a_bc += bce * posf;
    }

    float r;
    r = block_sum(a_np, sw); if (threadIdx.x == 0) partials[4 * blockIdx.x + 0] = r;
    r = block_sum(a_dw, sw); if (threadIdx.x == 0) partials[4 * blockIdx.x + 1] = r;
    r = block_sum(a_w,  sw); if (threadIdx.x == 0) partials[4 * blockIdx.x + 2] = r;
    r = block_sum(a_bc, sw); if (threadIdx.x == 0) partials[4 * blockIdx.x + 3] = r;
}

// ---------------------------------------------------------------------------
// Kernel D: single-block final combine.
// ---------------------------------------------------------------------------
__global__ void final_kernel(const float*  __restrict__ fpart, int nf,
                             const float4* __restrict__ lpart4, int nl,
                             float* __restrict__ out) {
    __shared__ float sw[8];
    float f = 0.0f, np = 0.0f, dw = 0.0f, wsum = 0.0f, bc = 0.0f;
    for (int i = threadIdx.x; i < nf; i += blockDim.x) f += fpart[i];
    for (int i = threadIdx.x; i < nl; i += blockDim.x) {
        const float4 lp = lpart4[i];
        np += lp.x; dw += lp.y; wsum += lp.z; bc += lp.w;
    }
    f    = block_sum(f,    sw);
    np   = block_sum(np,   sw);
    dw   = block_sum(dw,   sw);
    wsum = block_sum(wsum, sw);
    bc   = block_sum(bc,   sw);
    if (threadIdx.x == 0) {
        const float conf_loss   = f / (np + (float)BATCH);
        const float loc_loss    = dw / fmaxf(wsum, EPSV);
        const float center_loss = bc / fmaxf(np, 1.0f);
        out[0] = conf_loss + loc_loss + center_loss;
    }
}

// ---------------------------------------------------------------------------
extern "C" void kernel_launch(void* const* d_in, const int* in_sizes, int n_in,
                              void* d_out, int out_size, void* d_ws, size_t ws_size,
                              hipStream_t stream) {
    const float* plocs  = (const float*)d_in[0];   // (B, L, 4)
    const float* scores = (const float*)d_in[1];   // (B, L, C)
    const float* pctr   = (const float*)d_in[2];   // (B, L)
    const float* boxes  = (const float*)d_in[3];   // (B, M, 4)
    const int*   labels = (const int*)  d_in[4];   // (B, M)
    const float* locs   = (const float*)d_in[5];   // (L, 2)
    float* out = (float*)d_out;

    // Workspace carve-up (all offsets 16B aligned).
    char* ws = (char*)d_ws;
    int*    lab_ws = (int*)ws;    ws += (size_t)BATCH * L_LOCS * sizeof(int);
    float4* bt_ws  = (float4*)ws; ws += (size_t)BATCH * L_LOCS * sizeof(float4);
    float*  fpart  = (float*)ws;  ws += (size_t)FOCAL_BLOCKS * sizeof(float);
    float*  lpart  = (float*)ws;

    const int nBL = BATCH * L_LOCS;

    dim3 gA((L_LOCS + 255) / 256, BATCH);
    assign_kernel<<<gA, 256, 0, stream>>>(boxes, labels, locs, lab_ws, bt_ws);

    focal_kernel<<<FOCAL_BLOCKS, 256, 0, stream>>>(
        (const float4*)scores, lab_ws, fpart, nBL * (NCLS / 4));

    loc_kernel<<<LOC_BLOCKS, 256, 0, stream>>>(
        (const float4*)plocs, pctr, lab_ws, bt_ws, lpart, nBL);

    final_kernel<<<1, 256, 0, stream>>>(
        fpart, FOCAL_BLOCKS, (const float4*)lpart, LOC_BLOCKS, out);
}